// HierarchicalRouter_48670569398518
// MI455X (gfx1250) — compile-verified
//
#include <hip/hip_runtime.h>

#define NG 16          // groups
#define EPG 16         // minis per group
#define TOPK 4
#define HIDDEN 8192
#define NTOK 8192
#define KC 512                 // K-chunk staged in LDS
#define LDS_STRIDE 516         // KC + 4 pad -> conflict-free b64 A reads
#define LOG_STRIDE 273         // 272 logits + 1 pad
#define BLOCK_THREADS 544      // 17 waves: 1 group tile + 16 mini tiles

typedef __attribute__((ext_vector_type(2))) float v2f;
typedef __attribute__((ext_vector_type(8))) float v8f;

__global__ __launch_bounds__(BLOCK_THREADS)
void router_fused_kernel(const float* __restrict__ hidden,
                         const float* __restrict__ gw,     // [16, 8192]
                         const float* __restrict__ mg,     // [16, 8192, 16]
                         float* __restrict__ out,          // [N*4 f32 | N*4 i32 | 1 f32]
                         float* __restrict__ partials)     // [gridDim.x * 32]
{
    __shared__ __align__(16) float smem[16 * LDS_STRIDE];  // 33 KB; reused for logits/probs
    const int tid  = threadIdx.x;
    const int lane = tid & 31;
    const int w    = tid >> 5;            // 0..16 : output 16-col tile
    const int tok0 = blockIdx.x * 16;

    const int nCol  = lane & 15;          // N (B/C column), also M for A
    const int kHalf = (lane >> 4) << 1;   // lanes 0-15 -> K0,K1 ; lanes 16-31 -> K2,K3

    v8f acc = {0.f, 0.f, 0.f, 0.f, 0.f, 0.f, 0.f, 0.f};
    const float* aLds = smem + (size_t)(lane & 15) * LDS_STRIDE + kHalf;

    for (int k0 = 0; k0 < HIDDEN; k0 += KC) {
        __syncthreads();
        // Cooperatively stage A tile: 16 tokens x KC cols of fp32 (float4 vectors)
        for (int idx = tid; idx < 16 * (KC / 4); idx += BLOCK_THREADS) {
            const int m  = idx >> 7;            // KC/4 = 128 float4 per row
            const int c4 = idx & 127;
            const float4 v = *(const float4*)(hidden + (size_t)(tok0 + m) * HIDDEN + k0 + (c4 << 2));
            *(float4*)(smem + (size_t)m * LDS_STRIDE + (c4 << 2)) = v;
        }
        __syncthreads();

        if (w == 0) {
            // group gate: W[h][g] = gw[g*H + h]  -> contiguous float2 per lane
            const float* bp = gw + (size_t)nCol * HIDDEN + k0 + kHalf;
            #pragma unroll 4
            for (int kk = 0; kk < KC; kk += 4) {
                v2f a = *(const v2f*)(aLds + kk);
                v2f b = *(const v2f*)(bp + kk);
                acc = __builtin_amdgcn_wmma_f32_16x16x4_f32(
                        false, a, false, b, (short)0, acc, false, false);
            }
        } else {
            // mini gate group g: W[h][e] = mg[(g*H + h)*16 + e] (stride 16 floats in K)
            const int g = w - 1;
            const float* bp = mg + ((size_t)g * HIDDEN + k0 + kHalf) * EPG + nCol;
            #pragma unroll 4
            for (int kk = 0; kk < KC; kk += 4) {
                v2f a = *(const v2f*)(aLds + kk);
                v2f b;
                b.x = bp[(size_t)kk * EPG];
                b.y = bp[(size_t)kk * EPG + EPG];
                acc = __builtin_amdgcn_wmma_f32_16x16x4_f32(
                        false, a, false, b, (short)0, acc, false, false);
            }
        }
    }

    // ---- spill C tiles (16x16 each) to LDS logits[16][272] ----
    __syncthreads();
    {
        float* logits = smem;
        const int mBase   = (lane >> 4) * 8;              // C layout: VGPR r -> M = r + 8*(lane>=16)
        const int colBase = (w == 0) ? nCol : (16 + (w - 1) * 16 + nCol);
        #pragma unroll
        for (int r = 0; r < 8; ++r)
            logits[(mBase + r) * LOG_STRIDE + colBase] = acc[r];
    }
    __syncthreads();

    // ---- fused per-token epilogue (16 tokens, one thread each) ----
    float* logits   = smem;
    float* probsBuf = smem + 4400;   // 16 tokens x 32 probs (after 16*273=4368 logits)
    if (tid < 16) {
        const int t = tid;
        // group softmax + argmax
        float gl[NG];
        float mx = -3.0e38f;
        for (int g = 0; g < NG; ++g) { gl[g] = logits[t * LOG_STRIDE + g]; mx = fmaxf(mx, gl[g]); }
        float s = 0.f;
        for (int g = 0; g < NG; ++g) { gl[g] = __expf(gl[g] - mx); s += gl[g]; }
        float inv = 1.f / s;
        int   gi = 0; float gp = -1.f;
        for (int g = 0; g < NG; ++g) { gl[g] *= inv; if (gl[g] > gp) { gp = gl[g]; gi = g; } }
        // selected group's mini softmax
        float ml[EPG];
        mx = -3.0e38f;
        for (int e = 0; e < EPG; ++e) { ml[e] = logits[t * LOG_STRIDE + 16 + gi * EPG + e]; mx = fmaxf(mx, ml[e]); }
        s = 0.f;
        for (int e = 0; e < EPG; ++e) { ml[e] = __expf(ml[e] - mx); s += ml[e]; }
        inv = 1.f / s;
        for (int e = 0; e < EPG; ++e) ml[e] *= inv;
        // top-4 (strict > keeps smaller index on ties, matching lax.top_k)
        float tmp[EPG];
        for (int e = 0; e < EPG; ++e) tmp[e] = ml[e];
        int   idxs[TOPK]; float vals[TOPK]; float fs = 0.f;
        for (int j = 0; j < TOPK; ++j) {
            int bi = 0; float bv = tmp[0];
            for (int e = 1; e < EPG; ++e) if (tmp[e] > bv) { bv = tmp[e]; bi = e; }
            idxs[j] = bi; vals[j] = gp * bv; fs += vals[j];
            tmp[bi] = -3.0e38f;
        }
        inv = 1.f / fs;
        float* fpOut = out;
        int*   idOut = (int*)out + (size_t)NTOK * TOPK;
        for (int j = 0; j < TOPK; ++j) {
            fpOut[(size_t)(tok0 + t) * TOPK + j] = vals[j] * inv;
            idOut[(size_t)(tok0 + t) * TOPK + j] = gi * EPG + idxs[j];
        }
        // stash probs for aux-loss reduction
        for (int g = 0; g < NG; ++g)  probsBuf[t * 32 + g]       = gl[g];
        for (int e = 0; e < EPG; ++e) probsBuf[t * 32 + 16 + e]  = ml[e];
    }
    __syncthreads();
    if (tid < 32) {
        float s = 0.f;
        for (int t = 0; t < 16; ++t) s += probsBuf[t * 32 + tid];
        partials[(size_t)blockIdx.x * 32 + tid] = s;   // fixed-order, deterministic
    }
}

__global__ void router_aux_kernel(const float* __restrict__ partials,
                                  float* __restrict__ out, int nBlocks)
{
    __shared__ float red[32];
    const int j = threadIdx.x;
    if (j < 32) {
        float s = 0.f;
        for (int b = 0; b < nBlocks; ++b) s += partials[(size_t)b * 32 + j];
        const float mean = s * (1.0f / (float)NTOK);
        red[j] = mean * mean;
    }
    __syncthreads();
    if (j == 0) {
        float t = 0.f;
        for (int k = 0; k < 32; ++k) t += red[k];
        out[(size_t)NTOK * TOPK * 2] = t;   // total_aux_loss
    }
}

extern "C" void kernel_launch(void* const* d_in, const int* in_sizes, int n_in,
                              void* d_out, int out_size, void* d_ws, size_t ws_size,
                              hipStream_t stream) {
    const float* hidden = (const float*)d_in[0];   // [8192, 8192] f32
    const float* gw     = (const float*)d_in[1];   // [16, 8192]   f32
    const float* mg     = (const float*)d_in[2];   // [16, 8192, 16] f32
    float* out      = (float*)d_out;               // 65537 elements
    float* partials = (float*)d_ws;                // 512*32 f32 = 64 KB

    const int nBlocks = NTOK / 16;                 // 512
    router_fused_kernel<<<nBlocks, BLOCK_THREADS, 0, stream>>>(hidden, gw, mg, out, partials);
    router_aux_kernel<<<1, 32, 0, stream>>>(partials, out, nBlocks);
}